// TokenMappingWrapper_67173288509571
// MI455X (gfx1250) — compile-verified
//
#include <hip/hip_runtime.h>

// Token remap + embedding gather:
//   mapped = clamp(lookup_table[input_ids], 0, PRUNED_VOCAB-1)
//   out    = emb_weight[mapped]              // [B*S, 768] f32
//
// Pure data movement (0 FLOPs) -> WMMA not applicable. MI455X plan:
//  - wave32: one wave copies one 3072B embedding row, 6x b128 per lane
//    (32 lanes x 16B = 512B contiguous per wavefront transaction)
//  - the token id / mapped row are wave-uniform -> readfirstlane them so the
//    id->lut gather chain runs on the scalar (SMEM/KMcnt) path and all row
//    loads/stores use SGPR-base + lane-offset addressing (minimal VALU)
//  - NT stores for the 201MB output so the 98MB emb table stays resident in
//    the 192MB L2 (steady-state HBM traffic ~= writes only)

typedef float v4f __attribute__((ext_vector_type(4)));

constexpr int kEmbDim       = 768;
constexpr int kVecPerRow    = kEmbDim / 4;          // 192 float4 per row
constexpr int kLanes        = 32;                   // wave32
constexpr int kVecPerLane   = kVecPerRow / kLanes;  // 6
constexpr int kWavesPerBlk  = 8;
constexpr int kBlockThreads = kWavesPerBlk * kLanes; // 256

__global__ __launch_bounds__(kBlockThreads)
void token_embed_gather_kernel(const int* __restrict__ input_ids,
                               const int* __restrict__ lut,
                               const v4f* __restrict__ emb,
                               v4f* __restrict__ out,
                               int n_tokens,
                               int vocab_minus1)
{
    const int wave_v = blockIdx.x * kWavesPerBlk + (threadIdx.x >> 5);
    const int lane   = threadIdx.x & 31;
    if (wave_v >= n_tokens) return;   // wave-uniform guard (1 wave = 1 token)

    // All 32 lanes of a wave agree on wave_v -> promote to SGPR so the
    // dependent id->lut->row chain is scalar (s_load / SALU), not VMEM/VALU.
    const int wave = __builtin_amdgcn_readfirstlane(wave_v);

    int tok    = input_ids[wave];   // uniform -> s_load_b32
    int mapped = lut[tok];          // uniform -> s_load_b32 (L2-resident 512KB)

    // Canonical max-then-min clamp: folds to med3 / s_max+s_min, no branches.
    mapped = (mapped < 0) ? 0 : mapped;
    mapped = (mapped > vocab_minus1) ? vocab_minus1 : mapped;
    const int row = __builtin_amdgcn_readfirstlane(mapped);

    const v4f* __restrict__ src = emb + (size_t)row  * kVecPerRow + lane;
    v4f* __restrict__       dst = out + (size_t)wave * kVecPerRow + lane;

    // 6 independent b128 loads (overlap under LOADcnt), then 6 NT b128
    // stores. RT loads keep emb rows hot in L2; NT stores keep the write
    // stream from evicting them.
    v4f r[kVecPerLane];
#pragma unroll
    for (int i = 0; i < kVecPerLane; ++i)
        r[i] = src[i * kLanes];
#pragma unroll
    for (int i = 0; i < kVecPerLane; ++i)
        __builtin_nontemporal_store(r[i], dst + i * kLanes);
}

extern "C" void kernel_launch(void* const* d_in, const int* in_sizes, int n_in,
                              void* d_out, int out_size, void* d_ws, size_t ws_size,
                              hipStream_t stream)
{
    const int*   input_ids = (const int*)d_in[0];   // [B*S] int32
    const int*   lut       = (const int*)d_in[1];   // [ORIG_VOCAB] int32
    const float* emb       = (const float*)d_in[2]; // [PRUNED_VOCAB, 768] f32
    float*       out       = (float*)d_out;         // [B*S, 768] f32

    const int n_tokens = in_sizes[0];
    const int vocab    = in_sizes[2] / kEmbDim;

    const int blocks = (n_tokens + kWavesPerBlk - 1) / kWavesPerBlk;
    token_embed_gather_kernel<<<blocks, kBlockThreads, 0, stream>>>(
        input_ids, lut, (const v4f*)emb, (v4f*)out, n_tokens, vocab - 1);
}